// HierarchicalDistanceLoss_55276229099810
// MI455X (gfx1250) — compile-verified
//
#include <hip/hip_runtime.h>
#include <math.h>

// ---------------------------------------------------------------------------
// HierarchicalDistanceLoss for MI455X (gfx1250, wave32)
//   B = 1048576 rows, C = 40 classes. Bandwidth-bound (~168 MB -> ~7.2us at
//   23.3 TB/s). Strategy: async global->LDS tile streaming (ASYNCcnt pipeline,
//   double buffered, row-padded during copy), 1 row per lane compute,
//   deterministic two-stage mean reduction.
// ---------------------------------------------------------------------------

#define C_CLS          40
#define PAD_DW         44                       // padded row stride in dwords (176B, 16B aligned, 2-way banks)
#define TILE_ROWS      32                       // one row per lane of a wave32
#define TILES_PER_WAVE 8
#define WAVES_PER_WG   4
#define THREADS        (WAVES_PER_WG * 32)      // 128
#define ROWS_PER_WAVE  (TILE_ROWS * TILES_PER_WAVE)   // 256
#define ROWS_PER_WG    (WAVES_PER_WG * ROWS_PER_WAVE) // 1024
#define CHUNKS_PER_TILE 10                      // 32 rows * 160B / (32 lanes * 16B)

#if defined(__has_builtin)
#if __has_builtin(__builtin_amdgcn_global_load_async_to_lds_b128)
#define HAVE_ASYNC_LDS 1
#endif
#endif
#ifndef HAVE_ASYNC_LDS
#define HAVE_ASYNC_LDS 0
#endif

// The builtin's parameters are GCC-vector int4 pointers in explicit address
// spaces: (v4i addrspace(1)* src, v4i addrspace(3)* dst, imm offset, imm cpol).
typedef int v4i __attribute__((vector_size(16)));
typedef __attribute__((address_space(1))) v4i gas_v4i;   // global
typedef __attribute__((address_space(3))) v4i las_v4i;   // LDS

__device__ __forceinline__ void async_copy_b128(const float* g, float* l) {
#if HAVE_ASYNC_LDS
  // global_load_async_to_lds_b128: per-lane 16B global -> LDS, tracked by ASYNCcnt
  __builtin_amdgcn_global_load_async_to_lds_b128(
      (gas_v4i*)(g), (las_v4i*)(l), /*offset=*/0, /*cpol=*/0);
#else
  // synchronous staging fallback (still coalesced b128 load + ds_store_b128)
  *(float4*)l = *(const float4*)g;
#endif
}

// Wait until at most N async loads remain outstanding. In-order completion of
// async loads means "<=10 left" == "everything before the last 10 is in LDS".
__device__ __forceinline__ void wait_async_le10() {
#if HAVE_ASYNC_LDS
  asm volatile("s_wait_asynccnt 10" ::: "memory");
#else
  asm volatile("s_wait_dscnt 0" ::: "memory");
#endif
}
__device__ __forceinline__ void wait_async_le0() {
#if HAVE_ASYNC_LDS
  asm volatile("s_wait_asynccnt 0" ::: "memory");
#else
  asm volatile("s_wait_dscnt 0" ::: "memory");
#endif
}

__global__ __launch_bounds__(THREADS)
void hdl_main(const float* __restrict__ logits,
              const int*   __restrict__ labels,
              const float* __restrict__ dmat,
              float*       __restrict__ df_out,    // d_out + 1, [B]
              float*       __restrict__ partials)  // d_ws, one per block
{
  // 4 waves * 2 buffers * 32 rows * 44dw = 45056 B, + 6400 B dis_matrix
  __shared__ float s_tile[WAVES_PER_WG][2][TILE_ROWS * PAD_DW] __attribute__((aligned(16)));
  __shared__ float s_dm[C_CLS * C_CLS];
  __shared__ float s_part[WAVES_PER_WG];

  const int  tid  = threadIdx.x;
  const int  wave = tid >> 5;
  const int  lane = tid & 31;
  const long wave_row0 = (long)blockIdx.x * ROWS_PER_WG + (long)wave * ROWS_PER_WAVE;
  const float* wave_src = logits + wave_row0 * C_CLS;

  // Kick off async prefetch of tile 0 immediately (wave-private LDS region,
  // no barrier needed; overlaps with the dis_matrix staging below).
  {
    const float* g = wave_src;
    float*       l = &s_tile[wave][0][0];
#pragma unroll
    for (int i = 0; i < CHUNKS_PER_TILE; ++i) {
      const int chunk = i * 32 + lane;          // 16B chunk index in tile
      const int row   = chunk / 10;             // 40 floats = 10 chunks per row
      const int cq    = chunk - row * 10;
      async_copy_b128(g + chunk * 4, l + row * PAD_DW + cq * 4);
    }
  }

  // Stage the 40x40 distance matrix into LDS (6.4 KB, L2-resident anyway).
  for (int i = tid; i < C_CLS * C_CLS; i += THREADS) s_dm[i] = dmat[i];
  __syncthreads();

  float acc = 0.0f;

  for (int t = 0; t < TILES_PER_WAVE; ++t) {
    const int buf = t & 1;

    if (t + 1 < TILES_PER_WAVE) {
      // Issue next tile into the other buffer, then wait for current tile.
      const float* g = wave_src + (long)(t + 1) * TILE_ROWS * C_CLS;
      float*       l = &s_tile[wave][buf ^ 1][0];
#pragma unroll
      for (int i = 0; i < CHUNKS_PER_TILE; ++i) {
        const int chunk = i * 32 + lane;
        const int row   = chunk / 10;
        const int cq    = chunk - row * 10;
        async_copy_b128(g + chunk * 4, l + row * PAD_DW + cq * 4);
      }
      wait_async_le10();   // current tile's 10 loads done; next 10 in flight
    } else {
      wait_async_le0();
    }

    const float* rowp = &s_tile[wave][buf][lane * PAD_DW];

    // Pass 1: row max + first-occurrence argmax (matches jnp.argmax).
    float m  = -3.4028234663852886e38f;
    int   am = 0;
#pragma unroll
    for (int j = 0; j < 10; ++j) {
      const float4 q = *(const float4*)(rowp + 4 * j);
      const int b = 4 * j;
      if (q.x > m) { m = q.x; am = b;     }
      if (q.y > m) { m = q.y; am = b + 1; }
      if (q.z > m) { m = q.z; am = b + 2; }
      if (q.w > m) { m = q.w; am = b + 3; }
    }

    // Pass 2: sum of exp(x - max)  (v_exp_f32).
    float s = 0.0f;
#pragma unroll
    for (int j = 0; j < 10; ++j) {
      const float4 q = *(const float4*)(rowp + 4 * j);
      s += __expf(q.x - m) + __expf(q.y - m) + __expf(q.z - m) + __expf(q.w - m);
    }

    const long r   = wave_row0 + (long)t * TILE_ROWS + lane;
    const int  lab = labels[r];                         // coalesced 32-int read
    const float ce = __logf(s) + m - rowp[lab];         // -log_softmax[label]
    const float df = s_dm[lab * C_CLS + am] + 0.5f;     // dis[gold,pred]+alpha
    df_out[r] = df;                                     // coalesced store
    acc += ce * df;
  }

  // wave32 butterfly reduction (ds_swizzle/permlane path)
#pragma unroll
  for (int off = 16; off > 0; off >>= 1) acc += __shfl_xor(acc, off, 32);
  if (lane == 0) s_part[wave] = acc;
  __syncthreads();
  if (tid == 0) {
    float x = 0.0f;
#pragma unroll
    for (int w = 0; w < WAVES_PER_WG; ++w) x += s_part[w];
    partials[blockIdx.x] = x;
  }
}

// Deterministic final reduction: fixed-order sum of per-block partials.
__global__ __launch_bounds__(256)
void hdl_reduce(const float* __restrict__ partials, float* __restrict__ out,
                int n, float invB)
{
  __shared__ float s[256];
  float a = 0.0f;
  for (int i = threadIdx.x; i < n; i += 256) a += partials[i];
  s[threadIdx.x] = a;
  __syncthreads();
  for (int off = 128; off > 0; off >>= 1) {
    if ((int)threadIdx.x < off) s[threadIdx.x] += s[threadIdx.x + off];
    __syncthreads();
  }
  if (threadIdx.x == 0) out[0] = s[0] * invB;   // mean(ce*df)/NORMALISE
}

extern "C" void kernel_launch(void* const* d_in, const int* in_sizes, int n_in,
                              void* d_out, int out_size, void* d_ws, size_t ws_size,
                              hipStream_t stream) {
  const float* logits = (const float*)d_in[0];   // [B, 40] f32
  const int*   labels = (const int*)d_in[1];     // [B] i32
  const float* dmat   = (const float*)d_in[2];   // [40, 40] f32
  float* out = (float*)d_out;                    // [0]=loss, [1..B]=distance_factor
  float* partials = (float*)d_ws;

  const int B    = in_sizes[1];                  // 1048576
  const int grid = B / ROWS_PER_WG;              // 1024 blocks (B divisible)

  hdl_main<<<grid, THREADS, 0, stream>>>(logits, labels, dmat, out + 1, partials);
  hdl_reduce<<<1, 256, 0, stream>>>(partials, out, grid, 1.0f / (float)B);
}